// GraphSuperResolutionNet_54065048322455
// MI455X (gfx1250) — compile-verified
//
#include <hip/hip_runtime.h>
#include <cstdint>
#include <cstddef>

// Problem constants (fixed by the reference: B=32, H=W=256, SCALE=8)
#define NIMG 32
#define WDIM 256
#define HWN  65536                 // 256*256
#define NTOT (NIMG * HWN)          // 2,097,152 HR pixels
#define NLR  (NIMG * 1024)         // 32,768 LR pixels
#define NBLK 128                   // persistent workgroups (== NLR / TPB)
#define TPB  256
#define MAXITER 400

typedef float v2f __attribute__((ext_vector_type(2)));
typedef float v8f __attribute__((ext_vector_type(8)));

// ---------------------------------------------------------------------------
// Wave32 sum via V_WMMA_F32_16X16X4_F32 with an all-ones B matrix.
// D[m][n] = rowsum_m(A) for every n, independent of the A/B lane mappings;
// summing a lane's 8 D VGPRs gives Sum(rows 0..7) on lanes 0-15 and
// Sum(rows 8..15) on lanes 16-31; one xor-16 shuffle-add completes the sum.
// Requires EXEC == all ones (all call sites are fully convergent).
// ---------------------------------------------------------------------------
__device__ __forceinline__ float wave_sum_wmma(float v) {
  v2f a;  a.x = v;   a.y = 0.f;
  v2f bo; bo.x = 1.f; bo.y = 1.f;
  v8f c = {};
  v8f d = __builtin_amdgcn_wmma_f32_16x16x4_f32(
      /*neg_a=*/false, a, /*neg_b=*/false, bo,
      /*c_mod=*/(short)0, c, /*reuse_a=*/false, /*reuse_b=*/false);
  float s = ((d[0] + d[1]) + (d[2] + d[3])) + ((d[4] + d[5]) + (d[6] + d[7]));
  float t = __shfl_xor(s, 16, 32);   // combine the two half-wave sums
  return s + t;
}

// ---------------------------------------------------------------------------
// Grid-wide barrier (atomic ticket + generation), device scope.
// ---------------------------------------------------------------------------
__device__ __forceinline__ void grid_sync(int* cnt, int* gen) {
  __threadfence();            // release my writes to device scope
  __syncthreads();
  if (threadIdx.x == 0) {
    int g = __hip_atomic_load(gen, __ATOMIC_ACQUIRE, __HIP_MEMORY_SCOPE_AGENT);
    int t = __hip_atomic_fetch_add(cnt, 1, __ATOMIC_ACQ_REL, __HIP_MEMORY_SCOPE_AGENT);
    if (t == NBLK - 1) {
      __hip_atomic_store(cnt, 0, __ATOMIC_RELAXED, __HIP_MEMORY_SCOPE_AGENT);
      __hip_atomic_fetch_add(gen, 1, __ATOMIC_ACQ_REL, __HIP_MEMORY_SCOPE_AGENT);
    } else {
      while (__hip_atomic_load(gen, __ATOMIC_ACQUIRE, __HIP_MEMORY_SCOPE_AGENT) == g) {
        __builtin_amdgcn_s_sleep(1);
      }
    }
  }
  __syncthreads();
  __threadfence();            // acquire: invalidate stale near caches
}

// Deterministic grid-wide sum: WMMA wave sum -> 8 wave partials in LDS ->
// per-block partial -> every thread sums the 128 block partials in a fixed
// order (bitwise identical everywhere, every call, every replay).
__device__ float reduce_all(float v, float* partials, int* cnt, int* gen,
                            float* red) {
  float wsum = wave_sum_wmma(v);
  const int wave = threadIdx.x >> 5;
  const int lane = threadIdx.x & 31;
  if (lane == 0) red[wave] = wsum;
  __syncthreads();
  float blk = ((red[0] + red[1]) + (red[2] + red[3])) +
              ((red[4] + red[5]) + (red[6] + red[7]));
  if (threadIdx.x == 0) partials[blockIdx.x] = blk;
  grid_sync(cnt, gen);
  float t = 0.f;
  for (int j = 0; j < NBLK; ++j) t += partials[j];
  return t;
}

// ---------------------------------------------------------------------------
// Kernel 0: reset barrier state (ws is poisoned once, never re-poisoned).
// ---------------------------------------------------------------------------
__global__ void sync_init_kernel(int* s) { s[0] = 0; s[1] = 0; }

// ---------------------------------------------------------------------------
// Kernel 1: neighbor affinity -> written straight into the d_out aff region.
// aff layout [B,5,H,W]; channels already border-masked and scaled by lambda.
// ---------------------------------------------------------------------------
__global__ __launch_bounds__(TPB) void aff_kernel(
    const float* __restrict__ guide, const float* __restrict__ ybic,
    const float* __restrict__ log_lambda, const float* __restrict__ log_mu,
    float* __restrict__ aff) {
  int i = blockIdx.x * TPB + threadIdx.x;
  if (i >= NTOT) return;
  int b = i >> 16;
  int rem = i & 65535;
  int y = rem >> 8;
  int x = rem & 255;

  float mu  = __expf(log_mu[0]);
  float lam = __expf(log_lambda[0]);
  float inv = 1.0f / (1e-6f + mu * mu);

  const float* g0 = guide + (size_t)b * 3 * HWN + rem;
  float fc0 = g0[0], fc1 = g0[HWN], fc2 = g0[2 * HWN], fc3 = ybic[i];

  float at = 0.f, ab = 0.f, al = 0.f, ar = 0.f;
  if (y > 0) {
    float d0 = g0[-WDIM] - fc0, d1 = g0[HWN - WDIM] - fc1,
          d2 = g0[2 * HWN - WDIM] - fc2, d3 = ybic[i - WDIM] - fc3;
    at = __expf(-0.25f * (d0*d0 + d1*d1 + d2*d2 + d3*d3) * inv) * lam;
  }
  if (y < 255) {
    float d0 = g0[WDIM] - fc0, d1 = g0[HWN + WDIM] - fc1,
          d2 = g0[2 * HWN + WDIM] - fc2, d3 = ybic[i + WDIM] - fc3;
    ab = __expf(-0.25f * (d0*d0 + d1*d1 + d2*d2 + d3*d3) * inv) * lam;
  }
  if (x > 0) {
    float d0 = g0[-1] - fc0, d1 = g0[HWN - 1] - fc1,
          d2 = g0[2 * HWN - 1] - fc2, d3 = ybic[i - 1] - fc3;
    al = __expf(-0.25f * (d0*d0 + d1*d1 + d2*d2 + d3*d3) * inv) * lam;
  }
  if (x < 255) {
    float d0 = g0[1] - fc0, d1 = g0[HWN + 1] - fc1,
          d2 = g0[2 * HWN + 1] - fc2, d3 = ybic[i + 1] - fc3;
    ar = __expf(-0.25f * (d0*d0 + d1*d1 + d2*d2 + d3*d3) * inv) * lam;
  }
  float* ob = aff + (size_t)b * 5 * HWN + rem;
  ob[0]        = at;
  ob[HWN]      = ab;
  ob[2 * HWN]  = al;
  ob[3 * HWN]  = ar;
  ob[4 * HWN]  = at + ab + al + ar;
}

// ---------------------------------------------------------------------------
// Kernel 2: persistent cooperative CG. One thread <-> one 8x8 LR block.
// Per iteration: [stage p tile async->LDS; Ap + p.Ap] [reduce] [x,r update +
// r.r] [reduce] [p update + fused block-sum for pooling term] [sync].
// ---------------------------------------------------------------------------
__global__ __launch_bounds__(TPB) void cg_kernel(
    const float* __restrict__ source, const float* __restrict__ mask_lr,
    const float* __restrict__ aff, float* __restrict__ y_out,
    float* __restrict__ xv, float* __restrict__ rv, float* __restrict__ pv,
    float* __restrict__ apv, float* __restrict__ dsv,
    float* partA, float* partB, int* syncmem) {
  __shared__ float stage[66 * WDIM];   // 64-row tile + top/bottom halo
  __shared__ float red[8];             // one partial per wave

  int* cnt = syncmem;
  int* gen = syncmem + 1;

  const int tid = threadIdx.x;
  const int l   = blockIdx.x * TPB + tid;   // LR block id, [0, 32768)
  const int b   = l >> 10;                  // image
  const int q   = (l >> 8) & 3;             // 64-row strip within image
  const int lby = tid >> 5;                 // 0..7 block-row within strip
  const int by  = (l >> 5) & 31;
  const int bx  = l & 31;
  const int y0  = by << 3, x0 = bx << 3;
  const size_t img = (size_t)b << 16;
  const float* affb = aff + (size_t)b * 5 * HWN;

  // Warm L2 with this thread's affinity rows (global_prefetch_b8 path).
  #pragma unroll
  for (int c = 0; c < 5; ++c)
    __builtin_prefetch(affb + (size_t)c * HWN + (size_t)y0 * WDIM + x0, 0, 1);

  // ---- init: x=0, r=p=rhs, ds = mask*blocksum(p)/4096, bb = ||b||^2 ----
  const float mval = mask_lr[l];
  const float rv0  = mval * source[l] * (1.0f / 64.0f);   // rhs value (const per block)
  #pragma unroll
  for (int rr = 0; rr < 8; ++rr) {
    size_t base = img + (size_t)(y0 + rr) * WDIM + x0;
    #pragma unroll
    for (int cc = 0; cc < 8; ++cc) {
      xv[base + cc] = 0.f;
      rv[base + cc] = rv0;
      pv[base + cc] = rv0;
    }
  }
  dsv[l] = mval * rv0 * (1.0f / 64.0f);     // mask * (64*rv0) / 4096

  float bb = reduce_all(64.f * rv0 * rv0, partA, cnt, gen, red);
  float gamma = bb;
  const float atol2 = 1e-16f * bb;          // (tol*||b||)^2, tol=1e-8

  for (int it = 0; it < MAXITER; ++it) {
    if (!(gamma > atol2)) break;            // uniform across all threads

    // ---- stage p (66 x 256 halo tile) into LDS with async copies ----
    for (int idx = tid; idx < 66 * 64; idx += TPB) {
      int row = idx >> 6;                   // 0..65
      int c4  = (idx & 63) << 2;            // column (float4 granularity)
      int gy  = (q << 6) + row - 1;         // HR row in image
      float* lp = &stage[row * WDIM + c4];
      if ((unsigned)gy < 256u) {
        unsigned lds_off = (unsigned)(uintptr_t)lp;   // low 32b == LDS offset
        const float* gp = pv + img + (size_t)gy * WDIM + c4;
        asm volatile("global_load_async_to_lds_b128 %0, %1, off"
                     :: "v"(lds_off), "v"(gp) : "memory");
      } else {
        lp[0] = 0.f; lp[1] = 0.f; lp[2] = 0.f; lp[3] = 0.f;  // zero halo
      }
    }
    asm volatile("s_wait_asynccnt 0x0" ::: "memory");
    __syncthreads();

    // ---- Ap = L p + pooling coupling ; accumulate p.Ap ----
    const float dsl = dsv[l];
    float pq = 0.f;
    #pragma unroll
    for (int rr = 0; rr < 8; ++rr) {
      const int lr = (lby << 3) + rr + 1;   // LDS row for HR row y0+rr
      const size_t rem = (size_t)(y0 + rr) * WDIM + x0;
      #pragma unroll
      for (int cc = 0; cc < 8; ++cc) {
        const int gx = x0 + cc;
        float pc = stage[lr * WDIM + gx];
        float up = stage[(lr - 1) * WDIM + gx];
        float dn = stage[(lr + 1) * WDIM + gx];
        float lf = (gx > 0)   ? stage[lr * WDIM + gx - 1] : 0.f;
        float rt = (gx < 255) ? stage[lr * WDIM + gx + 1] : 0.f;
        float a0 = affb[rem + cc];
        float a1 = affb[HWN + rem + cc];
        float a2 = affb[2 * HWN + rem + cc];
        float a3 = affb[3 * HWN + rem + cc];
        float a4 = affb[4 * HWN + rem + cc];
        float ap = a4 * pc - (a0 * up + a1 * dn + a2 * lf + a3 * rt) + dsl;
        apv[img + rem + cc] = ap;
        pq += pc * ap;
      }
    }
    float pAp   = reduce_all(pq, partB, cnt, gen, red);
    float alpha = gamma / pAp;

    // ---- x += alpha p ; r -= alpha Ap ; accumulate r.r ----
    float g2 = 0.f;
    #pragma unroll
    for (int rr = 0; rr < 8; ++rr) {
      size_t base = img + (size_t)(y0 + rr) * WDIM + x0;
      #pragma unroll
      for (int cc = 0; cc < 8; ++cc) {
        size_t k = base + cc;
        xv[k] += alpha * pv[k];
        float rn = rv[k] - alpha * apv[k];
        rv[k] = rn;
        g2 += rn * rn;
      }
    }
    float rs_new = reduce_all(g2, partA, cnt, gen, red);
    float beta = rs_new / gamma;
    gamma = rs_new;

    // ---- p = r + beta p ; fused 8x8 block sum -> pooling term ----
    float s = 0.f;
    #pragma unroll
    for (int rr = 0; rr < 8; ++rr) {
      size_t base = img + (size_t)(y0 + rr) * WDIM + x0;
      #pragma unroll
      for (int cc = 0; cc < 8; ++cc) {
        size_t k = base + cc;
        float pn = rv[k] + beta * pv[k];
        pv[k] = pn;
        s += pn;
      }
    }
    dsv[l] = mval * s * (1.0f / 4096.0f);
    grid_sync(cnt, gen);                    // p/ds visible before next matvec
  }

  // ---- emit y_pred (this thread owns its pixels' x) ----
  #pragma unroll
  for (int rr = 0; rr < 8; ++rr) {
    size_t base = img + (size_t)(y0 + rr) * WDIM + x0;
    #pragma unroll
    for (int cc = 0; cc < 8; ++cc) y_out[base + cc] = xv[base + cc];
  }
}

// ---------------------------------------------------------------------------
extern "C" void kernel_launch(void* const* d_in, const int* in_sizes, int n_in,
                              void* d_out, int out_size, void* d_ws,
                              size_t ws_size, hipStream_t stream) {
  const float* guide      = (const float*)d_in[0];
  const float* y_bicubic  = (const float*)d_in[1];
  const float* source     = (const float*)d_in[2];
  const float* mask_lr    = (const float*)d_in[3];
  const float* log_lambda = (const float*)d_in[4];
  const float* log_mu     = (const float*)d_in[5];

  float* out   = (float*)d_out;
  float* y_out = out;              // [B,1,H,W]
  float* aff   = out + NTOT;       // [B,5,H,W]

  // Workspace layout: [sync ints | x | r | p | Ap | ds | partA | partB]
  char*  ws   = (char*)d_ws;
  int*   sy   = (int*)ws;
  float* f    = (float*)(ws + 256);
  float* xv   = f;
  float* rv   = f + (size_t)NTOT;
  float* pv   = f + 2 * (size_t)NTOT;
  float* apv  = f + 3 * (size_t)NTOT;
  float* dsv  = f + 4 * (size_t)NTOT;
  float* pa   = dsv + NLR;
  float* pb   = pa + NBLK;

  sync_init_kernel<<<1, 1, 0, stream>>>(sy);
  aff_kernel<<<NTOT / TPB, TPB, 0, stream>>>(guide, y_bicubic, log_lambda,
                                             log_mu, aff);
  cg_kernel<<<NBLK, TPB, 0, stream>>>(source, mask_lr, aff, y_out, xv, rv, pv,
                                      apv, dsv, pa, pb, sy);
}